// DCTConvLayer_39049842655607
// MI455X (gfx1250) — compile-verified
//
#include <hip/hip_runtime.h>
#include <hip/hip_bf16.h>

typedef __attribute__((ext_vector_type(2))) float v2f;
typedef __attribute__((ext_vector_type(4))) float v4f;
typedef __attribute__((ext_vector_type(8))) float v8f;

#define WIN     7
#define IH      384
#define IW      384
#define OH      377
#define OW      377
#define NBATCH  32
#define TI      16            // output rows per block
#define TJ      64            // output cols per block
#define XROWS   (TI + WIN - 1)   // 22
#define XCOLS   (TJ + WIN - 1)   // 70
#define XPITCH  72

// LDS pool layout (floats). Xs (pass-1 input patch) overlays the output
// staging buffers (pass-2) -- they are never live at the same time.
#define D_OFF   0                       // 49 (+3 pad) DCT coefficients
#define TS_OFF  52                      // Ts[22][7][64] = 9856
#define XS_OFF  (52 + 9856)             // Xs[22][72]    = 1584 (overlaid)
#define HF_OFF  (52 + 9856)             // SgHF[2][64][48] = 6144
#define ZF_REL  6144                    // SgZF = SgHF + 6144, 128 floats
#define POOL_SZ (52 + 9856 + 6144 + 128) // 16180 floats = 64720 B < 64 KB

__global__ __launch_bounds__(256)
void dct49_fused(const float* __restrict__ X,
                 float* __restrict__ zf,
                 float* __restrict__ hf)
{
    __shared__ __align__(16) float pool[POOL_SZ];
    float* Dl   = pool + D_OFF;
    float* Ts   = pool + TS_OFF;
    float* Xs   = pool + XS_OFF;
    float* SgHF = pool + HF_OFF;    // SgZF == SgHF + ZF_REL

    const int tid  = threadIdx.x;
    const int lane = tid & 31;
    const int wave = tid >> 5;
    const int half = lane >> 4;     // lane group (K split for WMMA frags)
    const int nidx = lane & 15;     // M index for A, N index for B/C

    const int b  = blockIdx.z;
    const int i0 = blockIdx.y * TI;
    const int j0 = blockIdx.x * TJ;

    // DCT matrix: D[k][m] = 2*cos(pi*k*(2m+1)/14)
    if (tid < 49) {
        int k = tid / 7, m = tid - 7 * (tid / 7);
        Dl[tid] = 2.0f * cosf(3.14159265358979323846f *
                              (float)(k * (2 * m + 1)) * (1.0f / 14.0f));
    }

    // ---- stage input patch (edge-clamped; clamped lanes feed only
    //      positions that are masked off at store time) ----
    const float* Xb = X + (size_t)b * IH * IW;
    for (int e = tid; e < XROWS * XCOLS; e += 256) {
        int r = e / XCOLS, c = e - r * XCOLS;
        int ic = i0 + r; if (ic > IH - 1) ic = IH - 1;
        int jc = j0 + c; if (jc > IW - 1) jc = IW - 1;
        Xs[r * XPITCH + c] = Xb[ic * IW + jc];
    }
    __syncthreads();

    // ---- constant B fragments: B[m,k] = D[k][m], zero-padded (m=7, k>=7).
    // Shared by BOTH DCT passes (B = D^T either way).
    // 16x4 f32 fragment layout: element(K) -> vgpr = K%2, lane-half = K/2
    v2f B0, B1;
    {
        int k = nidx;
        int m00 = 2 * half, m01 = 2 * half + 1;
        int m10 = 4 + 2 * half, m11 = 5 + 2 * half;
        B0.x = (k < 7) ? Dl[k * 7 + m00] : 0.0f;
        B0.y = (k < 7) ? Dl[k * 7 + m01] : 0.0f;
        B1.x = (k < 7) ? Dl[k * 7 + m10] : 0.0f;
        B1.y = (k < 7 && m11 < 7) ? Dl[k * 7 + m11] : 0.0f;  // m=7 pad
    }
    const int mlast = (half == 1) ? 6 : 5;   // K slot 5+2h, clamped (B=0)

    // ---- pass 1: horizontal DCT via WMMA ----
    // C[jj, l] = sum_n Xs[ii][jj+n] * D[l][n]; 88 tiles over 8 waves
    for (int t = wave; t < XROWS * 4; t += 8) {
        int ii  = t >> 2;
        int jg2 = t & 3;
        const float* xa = Xs + ii * XPITCH + jg2 * 16 + nidx;
        v2f A0, A1;
        A0.x = xa[2 * half];
        A0.y = xa[2 * half + 1];
        A1.x = xa[4 + 2 * half];
        A1.y = xa[mlast];
        v8f acc = {0.f, 0.f, 0.f, 0.f, 0.f, 0.f, 0.f, 0.f};
        acc = __builtin_amdgcn_wmma_f32_16x16x4_f32(
                  false, A0, false, B0, (short)0, acc, false, false);
        acc = __builtin_amdgcn_wmma_f32_16x16x4_f32(
                  false, A1, false, B1, (short)0, acc, false, false);
        // C: lane -> N=nidx (=l), vgpr r -> M = r + 8*half (=jj)
        if (nidx < 7) {
            float* tr = Ts + (ii * 7 + nidx) * TJ + jg2 * 16 + 8 * half;
            v4f lo = {acc[0], acc[1], acc[2], acc[3]};
            v4f hi = {acc[4], acc[5], acc[6], acc[7]};
            *(v4f*)tr       = lo;
            *(v4f*)(tr + 4) = hi;
        }
    }
    __syncthreads();   // Ts ready; Xs dead (staging buffers may overwrite)

    // ---- pass 2: vertical DCT via WMMA + staged streaming stores ----
    const int jg   = wave & 3;           // 16-col group handled by this wave
    const int irw  = wave >> 2;          // which row of the staged pair
    const int colA = jg * 16 + nidx;     // Ts column / j_local

    for (int ip = 0; ip < 8; ++ip) {
        const int il = 2 * ip + irw;                  // output row in tile
        const float* tbase = Ts + il * (WIN * TJ) + colA;

#pragma unroll
        for (int l = 0; l < 7; ++l) {
            const float* ta = tbase + l * TJ;
            // A[j,m] = T[il+m, colA, l]; 16x4 layout, m=7 clamped (B=0)
            v2f A0, A1;
            A0.x = ta[(2 * half    ) * (WIN * TJ)];
            A0.y = ta[(2 * half + 1) * (WIN * TJ)];
            A1.x = ta[(4 + 2 * half) * (WIN * TJ)];
            A1.y = ta[mlast * (WIN * TJ)];

            v8f acc = {0.f, 0.f, 0.f, 0.f, 0.f, 0.f, 0.f, 0.f};
            acc = __builtin_amdgcn_wmma_f32_16x16x4_f32(
                      false, A0, false, B0, (short)0, acc, false, false);
            acc = __builtin_amdgcn_wmma_f32_16x16x4_f32(
                      false, A1, false, B1, (short)0, acc, false, false);

            // C: lane -> N=nidx (=k), vgpr r -> M = r + 8*half (=j).
            // Branch-free select between hf staging and zf staging.
            if (nidx < 7) {
                const int o    = nidx * 7 + l;
                const int rowb = irw * TJ + jg * 16 + 8 * half;
#pragma unroll
                for (int r = 0; r < 8; ++r) {
                    int jlo = rowb + r;
                    int off = (o == 0) ? (ZF_REL + jlo)
                                       : (jlo * 48 + (o - 1));
                    SgHF[off] = acc[r];
                }
            }
        }
        __syncthreads();

        // ---- coalesced non-temporal streaming stores (b128) ----
#pragma unroll
        for (int q = 0; q < (2 * TJ * 48) / (4 * 256); ++q) {   // 6 iters
            int f   = (q * 256 + tid) * 4;
            int irr = f / (TJ * 48);
            int rem = f - irr * (TJ * 48);
            int jl  = rem / 48;
            int c0  = rem - jl * 48;
            int ig  = i0 + 2 * ip + irr;
            int jgl = j0 + jl;
            if (ig < OH && jgl < OW) {
                v4f v = *(const v4f*)(SgHF + f);
                size_t pos = ((size_t)b * OH + ig) * OW + jgl;
                __builtin_nontemporal_store(v, (v4f*)(hf + pos * 48 + c0));
            }
        }
        if (tid < 2 * TJ) {
            int irr = tid >> 6;
            int jl  = tid & 63;
            int ig  = i0 + 2 * ip + irr;
            int jgl = j0 + jl;
            if (ig < OH && jgl < OW)
                __builtin_nontemporal_store(
                    SgHF[ZF_REL + tid],
                    zf + ((size_t)b * OH + ig) * OW + jgl);
        }
        __syncthreads();
    }
}

extern "C" void kernel_launch(void* const* d_in, const int* in_sizes, int n_in,
                              void* d_out, int out_size, void* d_ws, size_t ws_size,
                              hipStream_t stream)
{
    (void)in_sizes; (void)n_in; (void)out_size; (void)d_ws; (void)ws_size;
    const float* X = (const float*)d_in[0];
    float* out = (float*)d_out;
    float* zf = out;                                   // 32*377*377
    float* hf = out + (size_t)NBATCH * OH * OW;        // 32*377*377*48
    dim3 grid((OW + TJ - 1) / TJ,   // 6
              (OH + TI - 1) / TI,   // 24
              NBATCH);              // 32
    dct49_fused<<<grid, 256, 0, stream>>>(X, zf, hf);
}